// MultiBoxLoss_70669391889253
// MI455X (gfx1250) — compile-verified
//
#include <hip/hip_runtime.h>
#include <math.h>

#define ALPHA_F 0.1f
#define OVERLAP_T 0.5f
#define NEG_POS_RATIO 3

typedef float v2f __attribute__((ext_vector_type(2)));
typedef float v8f __attribute__((ext_vector_type(8)));

// ---------------------------------------------------------------------------
// Wave-wide sum via V_WMMA_F32_16X16X4_F32 (verified lowering on gfx1250).
// A (16x4 f32): a.x = per-lane value, a.y = 0 -> row m of A holds
// {v_m, 0, v_(m+16), 0} in some K order; with B = all-ones, D[m][n] =
// v_m + v_(m+16) for every n (K-slot-mapping independent). C/D layout:
// VGPR j = row j (lanes 0-15) / row j+8 (lanes 16-31), so sum_j c[j] on lane 0
// covers rows 0-7 and on lane 16 rows 8-15. EXEC must be all ones.
// ---------------------------------------------------------------------------
__device__ __forceinline__ float wave_sum_wmma(float v) {
  v2f a; a.x = v;    a.y = 0.0f;
  v2f b; b.x = 1.0f; b.y = 1.0f;
  v8f c = {};
  c = __builtin_amdgcn_wmma_f32_16x16x4_f32(false, a, false, b, (short)0, c,
                                            false, false);
  float s = c[0] + c[1] + c[2] + c[3] + c[4] + c[5] + c[6] + c[7];
  return __shfl(s, 0, 32) + __shfl(s, 16, 32);
}

template <int NW>
__device__ __forceinline__ void block_atomic_add_f(float v, float* dst,
                                                   float* s_red) {
  float w = wave_sum_wmma(v);
  const int lane = threadIdx.x & 31, wid = threadIdx.x >> 5;
  if (lane == 0) s_red[wid] = w;
  __syncthreads();
  if (threadIdx.x == 0) {
    float t = 0.0f;
    for (int i = 0; i < NW; ++i) t += s_red[i];
    atomicAdd(dst, t);
  }
  __syncthreads();
}

__device__ __forceinline__ float smooth_l1(float d) {
  d = fabsf(d);
  return d < 1.0f ? 0.5f * d * d : d - 0.5f;
}

__device__ __forceinline__ float bce1(float x, float t) {
  return fmaxf(x, 0.0f) - x * t + log1pf(expf(-fabsf(x)));
}

// ---------------------------------------------------------------------------
// Kernel 1: matching + positive losses + loss_cpp production.
// One block per batch, 256 threads (8 waves). OT = compile-time cap on O so
// the per-truth argmax state (lv/lp) stays in VGPRs (no dynamic indexing).
// ---------------------------------------------------------------------------
template <int OT>
__global__ void __launch_bounds__(256) k_match(
    const float* __restrict__ loc, const float* __restrict__ cnf,
    const float* __restrict__ reg, const float* __restrict__ targets,
    const float* __restrict__ priors, float* __restrict__ wsLoss,
    unsigned char* __restrict__ wsConf, int* __restrict__ wsNpos,
    float* __restrict__ acc, int P, int O) {
  const int b = blockIdx.x;
  const int tid = threadIdx.x;

  __shared__ float s_t[OT * 6];
  __shared__ int s_bpi[OT];
  __shared__ float s_wv[8];
  __shared__ int s_wp[8];
  __shared__ float s_red[8];

  for (int i = tid; i < O * 6; i += 256) s_t[i] = targets[(size_t)b * O * 6 + i];
  __syncthreads();

  // ---- Pass A: one IoU sweep computes per-prior best truth (buffered to ws)
  // and per-truth best prior (kept in registers, reduced below).
  float lv[OT];
  int lp[OT];
#pragma unroll
  for (int o = 0; o < OT; ++o) { lv[o] = -1.0f; lp[o] = 0; }

  for (int p = tid; p < P; p += 256) {
    const float4 pr = ((const float4*)priors)[p];
    const float bx0 = pr.x - pr.z * 0.5f, by0 = pr.y - pr.w * 0.5f;
    const float bx1 = pr.x + pr.z * 0.5f, by1 = pr.y + pr.w * 0.5f;
    const float areaB = (bx1 - bx0) * (by1 - by0);
    float bo = -1.0f;
    int bi = 0;
#pragma unroll
    for (int o = 0; o < OT; ++o) {
      if (o < O) {
        const float ax0 = s_t[o * 6 + 0], ay0 = s_t[o * 6 + 1];
        const float ax1 = s_t[o * 6 + 2], ay1 = s_t[o * 6 + 3];
        const float areaA = (ax1 - ax0) * (ay1 - ay0);
        float ix = fminf(ax1, bx1) - fmaxf(ax0, bx0);
        float iy = fminf(ay1, by1) - fmaxf(ay0, by0);
        ix = fmaxf(ix, 0.0f);
        iy = fmaxf(iy, 0.0f);
        const float inter = ix * iy;
        const float iou = inter / (areaA + areaB - inter);
        if (iou > bo) { bo = iou; bi = o; }            // first-max over o
        if (iou > lv[o]) { lv[o] = iou; lp[o] = p; }   // first-max over p
      }
    }
    const size_t idx = (size_t)b * P + p;
    wsLoss[idx] = bo;                       // temp: best_truth_overlap
    wsConf[idx] = (unsigned char)bi;        // temp: best_truth_idx
  }

  // ---- Block argmax per truth (tie -> smaller prior index, JAX semantics).
#pragma unroll
  for (int o = 0; o < OT; ++o) {
    if (o < O) {
      float v = lv[o];
      int ip = lp[o];
      for (int off = 16; off; off >>= 1) {
        const float ov = __shfl_down(v, off, 32);
        const int oi = __shfl_down(ip, off, 32);
        if (ov > v || (ov == v && oi < ip)) { v = ov; ip = oi; }
      }
      if ((tid & 31) == 0) { s_wv[tid >> 5] = v; s_wp[tid >> 5] = ip; }
      __syncthreads();
      if (tid == 0) {
        float bv = s_wv[0];
        int bp = s_wp[0];
        for (int i = 1; i < 8; ++i)
          if (s_wv[i] > bv || (s_wv[i] == bv && s_wp[i] < bp)) {
            bv = s_wv[i]; bp = s_wp[i];
          }
        s_bpi[o] = bp;
      }
      __syncthreads();
    }
  }

  // ---- Pass B: overrides + fused positive losses + loss_cpp.
  float accl = 0.0f, accr = 0.0f;
  int np = 0;
  for (int p = tid; p < P; p += 256) {
    const size_t idx = (size_t)b * P + p;
    float bo = wsLoss[idx];               // same-thread RAW through global: ok
    int bi = (int)wsConf[idx];
#pragma unroll
    for (int o = 0; o < OT; ++o)
      if (o < O && s_bpi[o] == p) { bo = 2.0f; bi = o; }  // last write wins

    const int cls = (int)s_t[bi * 6 + 4];
    const int conf = (bo < OVERLAP_T) ? 0 : cls;
    const bool pos = conf > 0;

    const float4 pr = ((const float4*)priors)[p];
    const float mx0 = s_t[bi * 6 + 0], my0 = s_t[bi * 6 + 1];
    const float mx1 = s_t[bi * 6 + 2], my1 = s_t[bi * 6 + 3];
    const float lt0 = ((mx0 + mx1) * 0.5f - pr.x) / (0.1f * pr.z);
    const float lt1 = ((my0 + my1) * 0.5f - pr.y) / (0.1f * pr.w);

    const float2 l2 = ((const float2*)loc)[idx];
    const float rv = reg[idx];
    if (pos) {
      accl += smooth_l1(l2.x - lt0) + smooth_l1(l2.y - lt1);
      accr += smooth_l1(rv - s_t[bi * 6 + 5]);
      ++np;
    }

    if (p + 256 < P)
      __builtin_prefetch((const void*)(cnf + (idx + 256) * 4), 0, 1);
    const float4 x = ((const float4*)cnf)[idx];
    const float m = fmaxf(fmaxf(x.x, x.y), fmaxf(x.z, x.w));
    const float lse =
        m + logf(expf(x.x - m) + expf(x.y - m) + expf(x.z - m) + expf(x.w - m));
    const float picked = conf == 0 ? x.x : conf == 1 ? x.y : conf == 2 ? x.z : x.w;
    wsLoss[idx] = pos ? 0.0f : (lse - picked);  // >= 0 -> uint-monotonic key
    wsConf[idx] = (unsigned char)conf;
  }

  block_atomic_add_f<8>(accl, acc + 0, s_red);
  block_atomic_add_f<8>(accr, acc + 2, s_red);

  int w = np;
  for (int off = 16; off; off >>= 1) w += __shfl_down(w, off, 32);
  if ((tid & 31) == 0) s_wp[tid >> 5] = w;
  __syncthreads();
  if (tid == 0) {
    int t = 0;
    for (int i = 0; i < 8; ++i) t += s_wp[i];
    wsNpos[b] = t;
    atomicAdd(acc + 3, (float)t);
  }
}

// ---------------------------------------------------------------------------
// Kernel 2: exact hard-negative mining (LDS-resident radix select on the
// 128 KB key array -- fits in CDNA5's 320 KB WGP LDS) + fused BCE reduction.
// One block per batch, 1024 threads. The key staging HBM->LDS goes through
// the CDNA5 async pipe (GLOBAL_LOAD_ASYNC_TO_LDS_B128, ASYNCcnt): no VGPR
// round-trip, overlapped with the wsNpos scalar load, completed with an
// explicit s_wait_asynccnt before the workgroup barrier (the compiler's
// automatic waitcnt insertion does not track hand-issued async ops).
// ---------------------------------------------------------------------------
__global__ void __launch_bounds__(1024) k_mine(
    const float* __restrict__ wsLoss, const unsigned char* __restrict__ wsConf,
    const float* __restrict__ cnf, const int* __restrict__ wsNpos,
    float* __restrict__ acc, int P) {
  extern __shared__ unsigned int smem[];
  unsigned int* keys = smem;          // P words (dynamic LDS base: 16B aligned)
  unsigned int* mask = smem + P;      // P/32 words
  __shared__ int s_scan[1024];
  __shared__ int s_wred[32];
  __shared__ float s_fred[32];
  __shared__ int s_total;

  const int b = blockIdx.x, tid = threadIdx.x;
  const size_t base = (size_t)b * P;

  // Async HBM -> LDS: 4 keys (16B) per lane per issue; both addresses are
  // 16B aligned (P multiple of 4, keys at LDS dynamic base). Generic pointer
  // low 32 bits == LDS byte offset.
  {
    const int nvec = P >> 2;  // number of B128 transfers
    for (int i = tid; i < nvec; i += 1024) {
      const int p = i << 2;
      const unsigned int lds_off = (unsigned int)(uintptr_t)(keys + p);
      const float* g = wsLoss + base + p;
      asm volatile("global_load_async_to_lds_b128 %0, %1, off"
                   :: "v"(lds_off), "v"(g)
                   : "memory");
    }
  }
  const int num_pos = wsNpos[b];
  const int num_neg = min(NEG_POS_RATIO * num_pos, P - 1);
  asm volatile("s_wait_asynccnt 0x0" ::: "memory");
  __syncthreads();

  // Bitwise search: largest V with count(keys >= V) >= num_neg  ==>  V is the
  // num_neg-th largest key. Keys are non-negative floats -> bit-monotonic.
  unsigned int V = 0;
  int G = 0;
  if (num_neg > 0) {
    for (int bit = 31; bit >= 0; --bit) {
      const unsigned int cand = V | (1u << bit);
      int cnt = 0;
      for (int p = tid; p < P; p += 1024) cnt += (keys[p] >= cand) ? 1 : 0;
      for (int off = 16; off; off >>= 1) cnt += __shfl_down(cnt, off, 32);
      if ((tid & 31) == 0) s_wred[tid >> 5] = cnt;
      __syncthreads();
      if (tid == 0) {
        int t = 0;
        for (int i = 0; i < 32; ++i) t += s_wred[i];
        s_total = t;
      }
      __syncthreads();
      if (s_total >= num_neg) V = cand;  // identical decision on all threads
    }
    // G = count(keys > V)
    int cnt = 0;
    if (V != 0xFFFFFFFFu) {
      const unsigned int c2 = V + 1u;
      for (int p = tid; p < P; p += 1024) cnt += (keys[p] >= c2) ? 1 : 0;
    }
    for (int off = 16; off; off >>= 1) cnt += __shfl_down(cnt, off, 32);
    if ((tid & 31) == 0) s_wred[tid >> 5] = cnt;
    __syncthreads();
    if (tid == 0) {
      int t = 0;
      for (int i = 0; i < 32; ++i) t += s_wred[i];
      s_total = t;
    }
    __syncthreads();
    G = s_total;
  }
  const int R = num_neg - G;  // tie slots, assigned by ascending index

  // Exclusive prefix of tie counts over contiguous per-thread chunks
  // (matches stable descending argsort: ties resolved by ascending index).
  const int chunk = P / 1024;  // 32 for P=32768 (chunk%32==0 assumed)
  const int start = tid * chunk;
  int local = 0;
  if (num_neg > 0)
    for (int i = 0; i < chunk; ++i) local += (keys[start + i] == V) ? 1 : 0;
  s_scan[tid] = local;
  __syncthreads();
  for (int off = 1; off < 1024; off <<= 1) {
    const int v2 = (tid >= off) ? s_scan[tid - off] : 0;
    __syncthreads();
    s_scan[tid] += v2;
    __syncthreads();
  }
  int run = s_scan[tid] - local;

  for (int w = 0; w < chunk / 32; ++w) {
    unsigned int mw = 0;
    if (num_neg > 0) {
      for (int j = 0; j < 32; ++j) {
        const unsigned int k = keys[start + w * 32 + j];
        const bool tie = (k == V);
        const bool ng = (k > V) || (tie && run < R);
        run += tie ? 1 : 0;
        mw |= ng ? (1u << j) : 0u;
      }
    }
    mask[(start >> 5) + w] = mw;  // each thread owns its words: no atomics
  }
  __syncthreads();

  // Fused label-smoothed BCE over selected priors.
  float accc = 0.0f;
  for (int p = tid; p < P; p += 1024) {
    const int conf = (int)wsConf[base + p];
    const bool pos = conf > 0;
    const bool neg = (mask[p >> 5] >> (p & 31)) & 1u;
    const float sel = (pos || neg) ? 1.0f : 0.0f;
    if (p + 1024 < P)
      __builtin_prefetch((const void*)(cnf + (base + p + 1024) * 4), 0, 1);
    const float4 x = ((const float4*)cnf)[base + p];
    const float t0 = ALPHA_F * 0.25f + (conf == 0 ? (1.0f - ALPHA_F) : 0.0f);
    const float t1 = ALPHA_F * 0.25f + (conf == 1 ? (1.0f - ALPHA_F) : 0.0f);
    const float t2 = ALPHA_F * 0.25f + (conf == 2 ? (1.0f - ALPHA_F) : 0.0f);
    const float t3 = ALPHA_F * 0.25f + (conf == 3 ? (1.0f - ALPHA_F) : 0.0f);
    accc += sel * (bce1(x.x, t0) + bce1(x.y, t1) + bce1(x.z, t2) + bce1(x.w, t3));
  }
  block_atomic_add_f<32>(accc, acc + 1, s_fred);
}

__global__ void k_init(float* acc) {
  if (threadIdx.x < 4) acc[threadIdx.x] = 0.0f;
}

__global__ void k_final(const float* __restrict__ acc, float* __restrict__ out) {
  if (threadIdx.x == 0) {
    const float n = acc[3];
    out[0] = acc[0] / n;
    out[1] = acc[1] / n;
    out[2] = acc[2] / n;
  }
}

extern "C" void kernel_launch(void* const* d_in, const int* in_sizes, int n_in,
                              void* d_out, int out_size, void* d_ws,
                              size_t ws_size, hipStream_t stream) {
  const float* loc = (const float*)d_in[0];
  const float* cnf = (const float*)d_in[1];
  const float* reg = (const float*)d_in[2];
  const float* tgt = (const float*)d_in[3];
  const float* pri = (const float*)d_in[4];

  const int P = in_sizes[4] / 4;           // priors: P x 4
  const int B = in_sizes[2] / P;           // reg_data: B x P x 1
  const int O = in_sizes[3] / (6 * B);     // targets: B x O x 6

  char* ws = (char*)d_ws;
  const size_t nBP = (size_t)B * (size_t)P;
  float* wsLoss = (float*)ws;                               // B*P f32
  unsigned char* wsConf = (unsigned char*)(ws + nBP * 4);   // B*P u8
  size_t off = nBP * 5;
  off = (off + 255) & ~(size_t)255;
  int* wsNpos = (int*)(ws + off);                           // B i32
  float* accp = (float*)(ws + off + (((size_t)B * 4 + 15) & ~(size_t)15) + 16);

  k_init<<<1, 32, 0, stream>>>(accp);

  if (O <= 20) {
    k_match<20><<<B, 256, 0, stream>>>(loc, cnf, reg, tgt, pri, wsLoss, wsConf,
                                       wsNpos, accp, P, O);
  } else {
    k_match<32><<<B, 256, 0, stream>>>(loc, cnf, reg, tgt, pri, wsLoss, wsConf,
                                       wsNpos, accp, P, O);
  }

  const size_t smem2 = (size_t)P * 4 + (size_t)(P / 32) * 4;  // keys + mask
  k_mine<<<B, 1024, smem2, stream>>>(wsLoss, wsConf, cnf, wsNpos, accp, P);

  k_final<<<1, 32, 0, stream>>>(accp, (float*)d_out);
}